// SupervisedInstanceEmbeddingLoss_63058709840390
// MI455X (gfx1250) — compile-verified
//
#include <hip/hip_runtime.h>
#include <math.h>

typedef float v2f __attribute__((ext_vector_type(2)));
typedef float v8f __attribute__((ext_vector_type(8)));

#define BB    16
#define NPTS  16384
#define CCH   128
#define HH    512
#define WWID  512
#define NLAB  65
#define KLAB  64
#define CHUNK 512
#define EPSF  1e-12f

// ---------------- kernel 0: zero accumulators ----------------
__global__ void k_zero(float* __restrict__ counts, float* __restrict__ sums,
                       float* __restrict__ dsum, float* __restrict__ out) {
    int i = blockIdx.x * blockDim.x + threadIdx.x;
    if (i < BB * NLAB) { counts[i] = 0.0f; dsum[i] = 0.0f; }
    if (i < BB * NLAB * CCH) sums[i] = 0.0f;
    if (i == 0) out[0] = 0.0f;
}

// ---------------- kernel 1: gather labels + segment sums (LDS) ----------------
__global__ void __launch_bounds__(256)
k_label_sums(const float* __restrict__ emb, const int* __restrict__ coords,
             const int* __restrict__ y, int* __restrict__ labels,
             float* __restrict__ counts, float* __restrict__ sums) {
    __shared__ float s_sum[NLAB * CCH];   // 32.5 KB
    __shared__ float s_cnt[NLAB];
    __shared__ int   s_lab[CHUNK];

    const int b   = blockIdx.y;
    const int tid = threadIdx.x;
    const int base = b * NPTS + blockIdx.x * CHUNK;

    for (int i = tid; i < NLAB * CCH; i += 256) s_sum[i] = 0.0f;
    for (int i = tid; i < NLAB; i += 256) s_cnt[i] = 0.0f;
    __syncthreads();

    // gather labels for this chunk
    for (int p = tid; p < CHUNK; p += 256) {
        const int n = base + p;
        const int r = coords[2 * n + 0];
        const int c = coords[2 * n + 1];
        const int lab = y[b * (HH * WWID) + r * WWID + c];
        s_lab[p]  = lab;
        labels[n] = lab;
    }
    __syncthreads();

    // one wave per point: 32 lanes x float4 = 128 channels
    const int wave = tid >> 5;
    const int lane = tid & 31;
    for (int p = wave; p < CHUNK; p += 8) {
        const int lab = s_lab[p];
        const float4 e = ((const float4*)(emb + (size_t)(base + p) * CCH))[lane];
        float* dst = &s_sum[lab * CCH + lane * 4];
        atomicAdd(dst + 0, e.x);
        atomicAdd(dst + 1, e.y);
        atomicAdd(dst + 2, e.z);
        atomicAdd(dst + 3, e.w);
        if (lane == 0) atomicAdd(&s_cnt[lab], 1.0f);
    }
    __syncthreads();

    // merge to global
    float* gs = sums + (size_t)b * NLAB * CCH;
    for (int i = tid; i < NLAB * CCH; i += 256) atomicAdd(&gs[i], s_sum[i]);
    for (int i = tid; i < NLAB; i += 256)       atomicAdd(&counts[b * NLAB + i], s_cnt[i]);
}

// ---------------- kernel 2: centroids ----------------
__global__ void k_centroid(const float* __restrict__ sums, const float* __restrict__ counts,
                           float* __restrict__ cent) {
    int i = blockIdx.x * blockDim.x + threadIdx.x;
    if (i >= BB * NLAB * CCH) return;
    int bl = i / CCH;                       // b*NLAB + label
    cent[i] = sums[i] / fmaxf(counts[bl], 1.0f);
}

// ---------------- kernel 3: per-point distance, segment dist-sum ----------------
__global__ void __launch_bounds__(256)
k_dist(const float* __restrict__ emb, const int* __restrict__ labels,
       const float* __restrict__ cent, float* __restrict__ dsum) {
    __shared__ float s_cent[NLAB * CCH];   // 32.5 KB, reused by 512 points
    __shared__ float s_dist[NLAB];

    const int b   = blockIdx.y;
    const int tid = threadIdx.x;
    const int base = b * NPTS + blockIdx.x * CHUNK;

    const float* gc = cent + (size_t)b * NLAB * CCH;
    for (int i = tid; i < NLAB * CCH; i += 256) s_cent[i] = gc[i];
    for (int i = tid; i < NLAB; i += 256) s_dist[i] = 0.0f;
    __syncthreads();

    const int wave = tid >> 5;
    const int lane = tid & 31;
    for (int p = wave; p < CHUNK; p += 8) {
        const int n = base + p;
        const int lab = labels[n];
        const float4 e = ((const float4*)(emb + (size_t)n * CCH))[lane];
        const float4 c = ((const float4*)(&s_cent[lab * CCH]))[lane];
        float dx = e.x - c.x, dy = e.y - c.y, dz = e.z - c.z, dw = e.w - c.w;
        float sq = dx * dx + dy * dy + dz * dz + dw * dw;
        // wave32 reduction
        for (int off = 16; off > 0; off >>= 1) sq += __shfl_xor(sq, off, 32);
        if (lane == 0) {
            float d = sqrtf(fmaxf(sq, EPSF));
            atomicAdd(&s_dist[lab], d);
        }
    }
    __syncthreads();
    for (int i = tid; i < NLAB; i += 256) atomicAdd(&dsum[b * NLAB + i], s_dist[i]);
}

// ---------------- kernel 4: pull + WMMA f32 Gram -> push ----------------
__global__ void __launch_bounds__(512)
k_final(const float* __restrict__ cent, const float* __restrict__ counts,
        const float* __restrict__ dsum, float* __restrict__ out) {
    __shared__ float s_cent[KLAB * CCH];   // fg centroids, 32 KB
    __shared__ float s_gram[KLAB * KLAB];  // 16 KB
    __shared__ float s_cnt[NLAB];
    __shared__ float s_pull, s_push, s_npairs;

    const int b   = blockIdx.x;
    const int tid = threadIdx.x;

    for (int i = tid; i < NLAB; i += 512) s_cnt[i] = counts[b * NLAB + i];
    const float* gc = cent + (size_t)b * NLAB * CCH + CCH;     // skip label 0
    for (int i = tid; i < KLAB * CCH; i += 512) s_cent[i] = gc[i];
    if (tid == 0) { s_pull = 0.0f; s_push = 0.0f; s_npairs = 0.0f; }
    __syncthreads();

    // pull term: labels 1..64
    if (tid < KLAB) {
        float cnt = s_cnt[tid + 1];
        if (cnt > 0.0f) atomicAdd(&s_pull, dsum[b * NLAB + tid + 1] / cnt);
    }

    // Gram = C * C^T via V_WMMA_F32_16X16X4_F32; 16 waves -> 4x4 tiles of 16x16
    const int wave = tid >> 5;
    const int lane = tid & 31;
    const int ti = wave >> 2, tj = wave & 3;
    const int half = lane >> 4;            // 0: K=0,1   1: K=2,3
    const int l15  = lane & 15;
    const int arow = ti * 16 + l15;        // A: M = lane (mod 16)
    const int bcol = tj * 16 + l15;        // B: N = lane (mod 16)
    const int kofs = half * 2;

    v8f acc = {};
    #pragma unroll 4
    for (int k0 = 0; k0 < CCH; k0 += 4) {
        v2f a, bm;
        a.x  = s_cent[arow * CCH + k0 + kofs];
        a.y  = s_cent[arow * CCH + k0 + kofs + 1];
        bm.x = s_cent[bcol * CCH + k0 + kofs];
        bm.y = s_cent[bcol * CCH + k0 + kofs + 1];
        acc = __builtin_amdgcn_wmma_f32_16x16x4_f32(
                  false, a, false, bm, (short)0, acc, false, false);
    }
    // D layout: lane l15 -> N; VGPR r -> M = ti*16 + r + half*8
    #pragma unroll
    for (int r = 0; r < 8; ++r)
        s_gram[(ti * 16 + half * 8 + r) * KLAB + tj * 16 + l15] = acc[r];
    __syncthreads();

    // push term over upper-triangular present pairs
    float lp = 0.0f, ln = 0.0f;
    for (int idx = tid; idx < KLAB * KLAB; idx += 512) {
        const int i = idx >> 6, j = idx & 63;
        if (i < j && s_cnt[i + 1] > 0.0f && s_cnt[j + 1] > 0.0f) {
            float sq = s_gram[i * KLAB + i] + s_gram[j * KLAB + j]
                     - 2.0f * s_gram[i * KLAB + j];
            float dm = sqrtf(fmaxf(sq, EPSF));
            lp += fmaxf(1.0f - dm, 0.0f);
            ln += 1.0f;
        }
    }
    atomicAdd(&s_push, lp);
    atomicAdd(&s_npairs, ln);
    __syncthreads();

    if (tid == 0) {
        float push = s_push / fmaxf(s_npairs, 1.0f);
        atomicAdd(out, s_pull + push);
    }
}

// ---------------- launch ----------------
extern "C" void kernel_launch(void* const* d_in, const int* in_sizes, int n_in,
                              void* d_out, int out_size, void* d_ws, size_t ws_size,
                              hipStream_t stream) {
    const float* emb    = (const float*)d_in[0];
    const int*   coords = (const int*)d_in[1];
    const int*   y      = (const int*)d_in[2];
    float*       out    = (float*)d_out;

    char* ws = (char*)d_ws;
    int*   labels = (int*)ws;                                  // B*N ints
    float* counts = (float*)(ws + (size_t)BB * NPTS * 4);      // B*65
    float* sums   = counts + BB * NLAB;                        // B*65*128
    float* dsum   = sums + (size_t)BB * NLAB * CCH;            // B*65
    float* cent   = dsum + BB * NLAB;                          // B*65*128

    const int ztot = BB * NLAB * CCH;
    k_zero<<<(ztot + 255) / 256, 256, 0, stream>>>(counts, sums, dsum, out);

    dim3 g(NPTS / CHUNK, BB);
    k_label_sums<<<g, 256, 0, stream>>>(emb, coords, y, labels, counts, sums);
    k_centroid<<<(BB * NLAB * CCH + 255) / 256, 256, 0, stream>>>(sums, counts, cent);
    k_dist<<<g, 256, 0, stream>>>(emb, labels, cent, dsum);
    k_final<<<BB, 512, 0, stream>>>(cent, counts, dsum, out);
}